// RegionClassificationHead_50886772523205
// MI455X (gfx1250) — compile-verified
//
#include <hip/hip_runtime.h>
#include <hip/hip_bf16.h>
#include <math.h>

// ---------------- problem constants (from reference setup_inputs) ----------
#define NPROP 300
#define CIN   256
#define HF    50
#define WF    50
#define POOLP 7
#define FIN   (CIN * POOLP * POOLP)   // 12544
#define DHID  1024
#define NCLS  21
#define NREG  (NCLS * 4)              // 84
#define MROWS 320                     // M padded to 5 * 64
#define NCAND (NPROP * (NCLS - 1))    // 6000
#define NPAD  8192                    // bitonic pad
#define TOPKN 100
#define SCORE_THR 0.05f
#define NMS_THR   0.5f
#define MIN_SIZE  16.0f
#define KBIN  12
#define SMAX  16                      // max split-K slices (FC1)

typedef __attribute__((ext_vector_type(2))) float v2f;
typedef __attribute__((ext_vector_type(8))) float v8f;

// ---------------------------------------------------------------------------
// ROI max-pool (reference _roi_pool semantics), writes x[MROWS][FIN],
// zero-fills the padded rows 300..319.
// ---------------------------------------------------------------------------
__global__ __launch_bounds__(256) void roi_pool_kernel(
    const float* __restrict__ feat, const float* __restrict__ boxes,
    const int* __restrict__ canvas, float* __restrict__ x)
{
  const int n = blockIdx.x;
  if (n >= NPROP) {
    for (int e = threadIdx.x; e < FIN; e += blockDim.x)
      x[(size_t)n * FIN + e] = 0.0f;
    return;
  }
  __shared__ int s_hs[POOLP], s_hc[POOLP], s_ws[POOLP], s_wc[POOLP];
  if (threadIdx.x == 0) {
    const float scale = exp2f(rintf(log2f((float)HF / (float)canvas[0])));
    const float bx1 = rintf(boxes[n * 4 + 0] * scale);
    const float by1 = rintf(boxes[n * 4 + 1] * scale);
    const float bx2 = rintf(boxes[n * 4 + 2] * scale);
    const float by2 = rintf(boxes[n * 4 + 3] * scale);
    const float roi_w = fmaxf(bx2 - bx1 + 1.0f, 1.0f);
    const float roi_h = fmaxf(by2 - by1 + 1.0f, 1.0f);
    for (int p = 0; p < POOLP; p++) {
      float hs = fminf(fmaxf(floorf((float)p * roi_h / 7.0f) + by1, 0.0f), (float)HF);
      float he = fminf(fmaxf(ceilf((float)(p + 1) * roi_h / 7.0f) + by1, 0.0f), (float)HF);
      s_hs[p] = (int)hs;
      s_hc[p] = min(max((int)(he - hs), 0), KBIN);
      float wsv = fminf(fmaxf(floorf((float)p * roi_w / 7.0f) + bx1, 0.0f), (float)WF);
      float wev = fminf(fmaxf(ceilf((float)(p + 1) * roi_w / 7.0f) + bx1, 0.0f), (float)WF);
      s_ws[p] = (int)wsv;
      s_wc[p] = min(max((int)(wev - wsv), 0), KBIN);
    }
  }
  __syncthreads();
  for (int e = threadIdx.x; e < FIN; e += blockDim.x) {
    const int c  = e / (POOLP * POOLP);
    const int b  = e % (POOLP * POOLP);
    const int ph = b / POOLP, pw = b % POOLP;
    const int hs = s_hs[ph], hc = s_hc[ph];
    const int ws = s_ws[pw], wc = s_wc[pw];
    const float* fc = feat + (size_t)c * HF * WF;
    float m = -INFINITY;
    for (int kr = 0; kr < hc; kr++) {
      const int ri = min(hs + kr, HF - 1);
      const float* fr = fc + ri * WF;
      for (int kc = 0; kc < wc; kc++) {
        const int ci = min(ws + kc, WF - 1);
        m = fmaxf(m, fr[ci]);
      }
    }
    x[(size_t)n * FIN + e] = (hc > 0 && wc > 0) ? m : 0.0f;
  }
}

// ---------------------------------------------------------------------------
// Split-K fp32 WMMA GEMM partial:
//   Cpart[z][m][n] = sum_{k in slice z} A[m][k] * B[n][k]
// A: (MROWS x K) row-major (padded rows valid & zero)
// B: (N x K) row-major; rows clamped for N not multiple of 16
// grid = (N/16, MROWS/(16*MT), S). One wave per block; wave computes MT
// vertical 16x16 tiles of one N-tile, reusing streamed B registers MT times.
// Register double-buffered K loop keeps >=5 loads in flight per step so the
// matrix pipe is not serialized behind s_wait_loadcnt 0. Split-K multiplies
// wave count (FC1: 5120 waves) so aggregate outstanding loads can reach the
// 23.3 TB/s HBM roofline for the 51 MB W1 stream.
// A-layout (ISA 7.12.2, 32-bit A 16x4): lane<16: VGPR0=K0,VGPR1=K1;
// lane>=16: VGPR0=K2,VGPR1=K3 -> per-lane v2f of A[m][k+2*half .. +1].
// ---------------------------------------------------------------------------
template <int MT>
__global__ __launch_bounds__(32) void wmma_gemm_f32(
    const float* __restrict__ A, const float* __restrict__ B,
    float* __restrict__ Cpart, int N, int K, int kChunk)
{
  const int n0    = blockIdx.x << 4;
  const int m0    = blockIdx.y * (16 * MT);
  const int slice = blockIdx.z;
  const int k0    = slice * kChunk;
  const int lane  = threadIdx.x & 31;
  const int lm    = lane & 15;
  const int half  = lane >> 4;

  v8f acc[MT];
#pragma unroll
  for (int t = 0; t < MT; t++)
#pragma unroll
    for (int r = 0; r < 8; r++) acc[t][r] = 0.0f;

  int bn = n0 + lm;
  if (bn > N - 1) bn = N - 1;
  const float* __restrict__ bRow = B + (size_t)bn * K + k0 + (half << 1);
  const size_t aStep = (size_t)16 * (size_t)K;
  const float* __restrict__ aP[MT];
#pragma unroll
  for (int t = 0; t < MT; t++)
    aP[t] = A + (size_t)(m0 + lm) * K + k0 + (half << 1) + (size_t)t * aStep;

  // ---- software pipeline: prologue loads for k = 0 ----
  v2f bC = *(const v2f*)(bRow);
  v2f aC[MT];
#pragma unroll
  for (int t = 0; t < MT; t++) aC[t] = *(const v2f*)(aP[t]);

#pragma unroll 4
  for (int k = 0; k < kChunk - 4; k += 4) {
    // issue next-step loads before consuming current registers
    const v2f bN = *(const v2f*)(bRow + k + 4);
    v2f aN[MT];
#pragma unroll
    for (int t = 0; t < MT; t++) aN[t] = *(const v2f*)(aP[t] + k + 4);

#pragma unroll
    for (int t = 0; t < MT; t++) {
      acc[t] = __builtin_amdgcn_wmma_f32_16x16x4_f32(
          /*neg_a=*/false, aC[t], /*neg_b=*/false, bC,
          /*c_mod=*/(short)0, acc[t], /*reuse_a=*/false, /*reuse_b=*/false);
    }
    bC = bN;
#pragma unroll
    for (int t = 0; t < MT; t++) aC[t] = aN[t];
  }
  // ---- epilogue: last K-step ----
#pragma unroll
  for (int t = 0; t < MT; t++) {
    acc[t] = __builtin_amdgcn_wmma_f32_16x16x4_f32(
        false, aC[t], false, bC, (short)0, acc[t], false, false);
  }

  // C/D layout: VGPR r, lanes 0-15 -> M=r, lanes 16-31 -> M=r+8; N = lane&15
  const int n = n0 + lm;
  if (n < N) {
    float* __restrict__ out = Cpart + (size_t)slice * MROWS * N;
#pragma unroll
    for (int t = 0; t < MT; t++) {
      const int mb = m0 + t * 16 + (half << 3);
#pragma unroll
      for (int r = 0; r < 8; r++)
        out[(size_t)(mb + r) * N + n] = acc[t][r];
    }
  }
}

// ---------------------------------------------------------------------------
// Deterministic split-K reduction: C[m][n] = act( sum_z part[z][m][n] + b[n] )
// ---------------------------------------------------------------------------
__global__ __launch_bounds__(256) void reduce_bias_kernel(
    const float* __restrict__ part, const float* __restrict__ bias,
    float* __restrict__ C, int N, int S, int relu)
{
  const int i = blockIdx.x * blockDim.x + threadIdx.x;
  if (i >= MROWS * N) return;
  const int n = i % N;
  float s = 0.0f;
  for (int z = 0; z < S; z++) s += part[(size_t)z * MROWS * N + i];
  s += bias[n];
  if (relu) s = fmaxf(s, 0.0f);
  C[i] = s;
}

// ---------------------------------------------------------------------------
// Per-proposal softmax + box decode + clip + validity (classes 1..20)
// ---------------------------------------------------------------------------
__global__ __launch_bounds__(32) void decode_softmax_kernel(
    const float* __restrict__ logits, const float* __restrict__ regs,
    const float* __restrict__ props, const int* __restrict__ canvas,
    float* __restrict__ fb, float* __restrict__ fs,
    int* __restrict__ fl, int* __restrict__ fvalid)
{
  const int n = blockIdx.x;
  const int t = threadIdx.x;
  __shared__ float prob[NCLS];
  if (t < NCLS) prob[t] = logits[n * NCLS + t];
  __syncthreads();
  if (t == 0) {
    float mx = prob[0];
    for (int i = 1; i < NCLS; i++) mx = fmaxf(mx, prob[i]);
    float s = 0.0f;
    for (int i = 0; i < NCLS; i++) { float e = expf(prob[i] - mx); prob[i] = e; s += e; }
    const float inv = 1.0f / s;
    for (int i = 0; i < NCLS; i++) prob[i] *= inv;
  }
  __syncthreads();
  const float img_h = (float)canvas[0];
  const float img_w = (float)canvas[1];
  if (t >= 1 && t < NCLS) {
    const int c = t;
    const float px1 = props[n * 4 + 0], py1 = props[n * 4 + 1];
    const float px2 = props[n * 4 + 2], py2 = props[n * 4 + 3];
    const float w = px2 - px1, h = py2 - py1;
    const float cx = px1 + 0.5f * w, cy = py1 + 0.5f * h;
    const float dx = regs[n * NREG + c * 4 + 0];
    const float dy = regs[n * NREG + c * 4 + 1];
    const float dw = regs[n * NREG + c * 4 + 2];
    const float dh = regs[n * NREG + c * 4 + 3];
    const float pcx = dx * w + cx, pcy = dy * h + cy;
    const float pw = expf(dw) * w, ph = expf(dh) * h;
    const float bx1 = fminf(fmaxf(pcx - 0.5f * pw, 0.0f), img_w);
    const float by1 = fminf(fmaxf(pcy - 0.5f * ph, 0.0f), img_h);
    const float bx2 = fminf(fmaxf(pcx + 0.5f * pw, 0.0f), img_w);
    const float by2 = fminf(fmaxf(pcy + 0.5f * ph, 0.0f), img_h);
    const int j = n * (NCLS - 1) + (c - 1);
    fb[j * 4 + 0] = bx1; fb[j * 4 + 1] = by1;
    fb[j * 4 + 2] = bx2; fb[j * 4 + 3] = by2;
    const float sc = prob[c];
    fs[j] = sc;
    fl[j] = c;
    fvalid[j] = (sc > SCORE_THR) && ((bx2 - bx1) >= MIN_SIZE) && ((by2 - by1) >= MIN_SIZE);
  }
}

// ---------------------------------------------------------------------------
// Descending bitonic argsort of where(valid, fs, -1), stable tie-break by
// index (matches jnp.argsort(-key)). 8192 keys in LDS, single block.
// ---------------------------------------------------------------------------
__global__ __launch_bounds__(1024) void sort_kernel(
    const float* __restrict__ fs, const int* __restrict__ fvalid,
    int* __restrict__ sidx)
{
  extern __shared__ char smem_sort[];
  float* key = (float*)smem_sort;
  int* idx = (int*)(key + NPAD);
  for (int i = threadIdx.x; i < NPAD; i += blockDim.x) {
    key[i] = (i < NCAND) ? (fvalid[i] ? fs[i] : -1.0f) : -2.0f;
    idx[i] = i;
  }
  __syncthreads();
  for (int size = 2; size <= NPAD; size <<= 1) {
    for (int stride = size >> 1; stride > 0; stride >>= 1) {
      for (int t = threadIdx.x; t < NPAD / 2; t += blockDim.x) {
        const int i = 2 * stride * (t / stride) + (t % stride);
        const int j = i + stride;
        const bool dir = ((i & size) == 0);  // true -> descending segment
        const float ki = key[i], kj = key[j];
        const int xi = idx[i], xj = idx[j];
        const bool pre = (ki > kj) || (ki == kj && xi < xj);
        if (pre != dir) { key[i] = kj; key[j] = ki; idx[i] = xj; idx[j] = xi; }
      }
      __syncthreads();
    }
  }
  for (int i = threadIdx.x; i < NCAND; i += blockDim.x) sidx[i] = idx[i];
}

// ---------------------------------------------------------------------------
// Gather sorted arrays + class-offset boxes + initial keep flags
// ---------------------------------------------------------------------------
__global__ __launch_bounds__(256) void gather_kernel(
    const int* __restrict__ sidx, const float* __restrict__ fb,
    const float* __restrict__ fs, const int* __restrict__ fl,
    const int* __restrict__ fvalid,
    float* __restrict__ sb, float* __restrict__ ob, float* __restrict__ ss,
    int* __restrict__ sl, int* __restrict__ keep)
{
  const int i = blockIdx.x * blockDim.x + threadIdx.x;
  if (i >= NCAND) return;
  const int j = sidx[i];
  const float b0 = fb[j * 4 + 0], b1 = fb[j * 4 + 1];
  const float b2 = fb[j * 4 + 2], b3 = fb[j * 4 + 3];
  const int lab = fl[j];
  sb[i * 4 + 0] = b0; sb[i * 4 + 1] = b1; sb[i * 4 + 2] = b2; sb[i * 4 + 3] = b3;
  const float off = (float)lab * 10000.0f;
  ob[i * 4 + 0] = b0 + off; ob[i * 4 + 1] = b1 + off;
  ob[i * 4 + 2] = b2 + off; ob[i * 4 + 3] = b3 + off;
  ss[i] = fs[j];
  sl[i] = lab;
  keep[i] = fvalid[j];
}

// ---------------------------------------------------------------------------
// Greedy NMS, single block; boxes + keep flags live in LDS (120 KB dynamic,
// legal under CDNA5's 320 KB/WGP LDS).
// ---------------------------------------------------------------------------
__global__ __launch_bounds__(1024) void nms_kernel(
    const float* __restrict__ ob, int* __restrict__ keep)
{
  extern __shared__ char smem_nms[];
  float4* bx = (float4*)smem_nms;
  int* kp = (int*)(bx + NCAND);
  for (int i = threadIdx.x; i < NCAND; i += blockDim.x) {
    bx[i] = ((const float4*)ob)[i];
    kp[i] = keep[i];
  }
  __syncthreads();
  for (int i = 0; i < NCAND - 1; i++) {
    if (!kp[i]) continue;  // uniform across the block
    const float4 bi = bx[i];
    const float ai = fmaxf(bi.z - bi.x, 0.0f) * fmaxf(bi.w - bi.y, 0.0f);
    for (int j = i + 1 + threadIdx.x; j < NCAND; j += blockDim.x) {
      if (kp[j]) {
        const float4 bj = bx[j];
        const float aj = fmaxf(bj.z - bj.x, 0.0f) * fmaxf(bj.w - bj.y, 0.0f);
        const float lx = fmaxf(bi.x, bj.x), ly = fmaxf(bi.y, bj.y);
        const float rx = fminf(bi.z, bj.z), ry = fminf(bi.w, bj.w);
        const float inter = fmaxf(rx - lx, 0.0f) * fmaxf(ry - ly, 0.0f);
        const float uni = ai + aj - inter;
        if (inter / fmaxf(uni, 1e-9f) > NMS_THR) kp[j] = 0;
      }
    }
    __syncthreads();
  }
  for (int i = threadIdx.x; i < NCAND; i += blockDim.x) keep[i] = kp[i];
}

// ---------------------------------------------------------------------------
// Compact first TOPKN kept entries (already sorted desc) -> outputs
// d_out layout: boxes[100*4] f32, labels[100] i32 bits, scores[100] f32
// ---------------------------------------------------------------------------
__global__ void compact_kernel(
    const int* __restrict__ keep, const float* __restrict__ sb,
    const float* __restrict__ ss, const int* __restrict__ sl,
    float* __restrict__ out)
{
  if (threadIdx.x != 0 || blockIdx.x != 0) return;
  float* out_boxes  = out;
  int*   out_labels = (int*)(out + TOPKN * 4);
  float* out_scores = out + TOPKN * 5;
  int rank = 0;
  for (int i = 0; i < NCAND && rank < TOPKN; i++) {
    if (keep[i]) {
      out_boxes[rank * 4 + 0] = sb[i * 4 + 0];
      out_boxes[rank * 4 + 1] = sb[i * 4 + 1];
      out_boxes[rank * 4 + 2] = sb[i * 4 + 2];
      out_boxes[rank * 4 + 3] = sb[i * 4 + 3];
      out_labels[rank] = sl[i];
      out_scores[rank] = ss[i];
      rank++;
    }
  }
  for (; rank < TOPKN; rank++) {
    out_boxes[rank * 4 + 0] = 0.0f; out_boxes[rank * 4 + 1] = 0.0f;
    out_boxes[rank * 4 + 2] = 0.0f; out_boxes[rank * 4 + 3] = 0.0f;
    out_labels[rank] = 0;
    out_scores[rank] = 0.0f;
  }
}

// ---------------------------------------------------------------------------
extern "C" void kernel_launch(void* const* d_in, const int* in_sizes, int n_in,
                              void* d_out, int out_size, void* d_ws, size_t ws_size,
                              hipStream_t stream)
{
  const float* feat  = (const float*)d_in[0];
  const float* props = (const float*)d_in[1];
  const int*   canvas= (const int*)  d_in[2];
  const float* W1    = (const float*)d_in[3];
  const float* b1    = (const float*)d_in[4];
  const float* W2    = (const float*)d_in[5];
  const float* b2    = (const float*)d_in[6];
  const float* Wc    = (const float*)d_in[7];
  const float* bc    = (const float*)d_in[8];
  const float* Wr    = (const float*)d_in[9];
  const float* br    = (const float*)d_in[10];

  float* ws = (float*)d_ws;
  size_t off = 0;
  auto alloc = [&](size_t nelem) -> float* {
    float* p = ws + off;
    off += (nelem + 255) & ~(size_t)255;
    return p;
  };
  float* x      = alloc((size_t)MROWS * FIN);
  float* pbuf   = alloc((size_t)SMAX * MROWS * DHID);  // split-K partials
  float* h1     = alloc((size_t)MROWS * DHID);
  float* h2     = alloc((size_t)MROWS * DHID);
  float* logits = alloc((size_t)MROWS * NCLS);
  float* regs   = alloc((size_t)MROWS * NREG);
  float* fb     = alloc((size_t)NCAND * 4);
  float* fs     = alloc(NCAND);
  int*   fl     = (int*)alloc(NCAND);
  int*   fvalid = (int*)alloc(NCAND);
  int*   sidx   = (int*)alloc(NCAND);
  float* sb     = alloc((size_t)NCAND * 4);
  float* ob     = alloc((size_t)NCAND * 4);
  float* ss     = alloc(NCAND);
  int*   sl     = (int*)alloc(NCAND);
  int*   keep   = (int*)alloc(NCAND);
  if (off * sizeof(float) > ws_size) return;  // workspace too small: bail

  // 1. ROI max-pool -> x (padded rows zeroed)
  roi_pool_kernel<<<MROWS, 256, 0, stream>>>(feat, props, canvas, x);

  // 2. FC1: split-K x16 (5120 waves to saturate HBM on the 51MB W1 stream)
  wmma_gemm_f32<4><<<dim3(DHID / 16, MROWS / 64, 16), 32, 0, stream>>>(
      x, W1, pbuf, DHID, FIN, FIN / 16);
  reduce_bias_kernel<<<(MROWS * DHID + 255) / 256, 256, 0, stream>>>(
      pbuf, b1, h1, DHID, 16, 1);

  // 3. FC2: split-K x4
  wmma_gemm_f32<4><<<dim3(DHID / 16, MROWS / 64, 4), 32, 0, stream>>>(
      h1, W2, pbuf, DHID, DHID, DHID / 4);
  reduce_bias_kernel<<<(MROWS * DHID + 255) / 256, 256, 0, stream>>>(
      pbuf, b2, h2, DHID, 4, 1);

  // 4. classification logits (N=21 -> 2 N-tiles, clamped), S=1
  wmma_gemm_f32<4><<<dim3(2, MROWS / 64, 1), 32, 0, stream>>>(
      h2, Wc, pbuf, NCLS, DHID, DHID);
  reduce_bias_kernel<<<(MROWS * NCLS + 255) / 256, 256, 0, stream>>>(
      pbuf, bc, logits, NCLS, 1, 0);

  // 5. regression deltas (N=84 -> 6 N-tiles), S=1
  wmma_gemm_f32<4><<<dim3(6, MROWS / 64, 1), 32, 0, stream>>>(
      h2, Wr, pbuf, NREG, DHID, DHID);
  reduce_bias_kernel<<<(MROWS * NREG + 255) / 256, 256, 0, stream>>>(
      pbuf, br, regs, NREG, 1, 0);

  // 6. softmax + decode + clip + validity
  decode_softmax_kernel<<<NPROP, 32, 0, stream>>>(
      logits, regs, props, canvas, fb, fs, fl, fvalid);

  // 7. stable descending sort of 6000 candidates (bitonic, LDS)
  sort_kernel<<<1, 1024, NPAD * (sizeof(float) + sizeof(int)), stream>>>(
      fs, fvalid, sidx);

  // 8. gather sorted views + offset boxes
  gather_kernel<<<(NCAND + 255) / 256, 256, 0, stream>>>(
      sidx, fb, fs, fl, fvalid, sb, ob, ss, sl, keep);

  // 9. greedy NMS (LDS-resident boxes)
  nms_kernel<<<1, 1024, NCAND * sizeof(float4) + NCAND * sizeof(int), stream>>>(
      ob, keep);

  // 10. top-100 compaction -> d_out
  compact_kernel<<<1, 32, 0, stream>>>(keep, sb, ss, sl, (float*)d_out);
}